// GraphEncoderWithNodes_47244640256180
// MI455X (gfx1250) — compile-verified
//
#include <hip/hip_runtime.h>
#include <hip/hip_bf16.h>
#include <math.h>

// ---------------- CDNA5 WMMA fp32 16x16x4 ----------------
typedef __attribute__((ext_vector_type(2))) float v2f;
typedef __attribute__((ext_vector_type(8))) float v8f;

__device__ __forceinline__ v8f wmma4(v2f a, v2f b, v8f c) {
  // D = A(16x4) * B(4x16) + C(16x16), fp32
  return __builtin_amdgcn_wmma_f32_16x16x4_f32(false, a, false, b, (short)0, c,
                                               false, false);
}

__device__ __forceinline__ void atomicMaxF(float* addr, float val) {
  // signed-int max for non-negative floats, unsigned-int min for negative:
  // together implement float atomic-max for any mix of signs.
  if (val >= 0.0f)
    atomicMax((int*)addr, __float_as_int(val));
  else
    atomicMin((unsigned int*)addr, __float_as_uint(val));
}

// ---------------- generic WMMA GEMM: C = A(MxK) @ B(KxN) + bias ----------------
__global__ void gemm_bias_kernel(const float* __restrict__ A,
                                 const float* __restrict__ B,
                                 const float* __restrict__ bias,
                                 float* __restrict__ C,
                                 int M, int K, int Ncols) {
  const int lane = threadIdx.x & 31;
  const int half = lane >> 4;   // selects K pair (A) / K rows (B)
  const int idx  = lane & 15;   // M (A) / N (B,C)
  const int rowTiles = M >> 4, colTiles = Ncols >> 4;
  const int totalTiles = rowTiles * colTiles;
  const int nWaves = (gridDim.x * blockDim.x) >> 5;
  for (int wave = (blockIdx.x * blockDim.x + threadIdx.x) >> 5; wave < totalTiles;
       wave += nWaves) {
    const int rt = wave / colTiles, ct = wave % colTiles;
    const int c0 = ct * 16;
    const float* arow = A + (size_t)(rt * 16 + idx) * K;
    v8f acc = {};
    for (int k0 = 0; k0 < K; k0 += 4) {
      const int ka = k0 + 2 * half;
      v2f a, b;
      a[0] = arow[ka];
      a[1] = arow[ka + 1];
      b[0] = B[(size_t)ka * Ncols + c0 + idx];
      b[1] = B[(size_t)(ka + 1) * Ncols + c0 + idx];
      acc = wmma4(a, b, acc);
    }
    const float bv = bias ? bias[c0 + idx] : 0.0f;
#pragma unroll
    for (int r = 0; r < 8; ++r) {
      const int m = rt * 16 + r + 8 * half;
      C[(size_t)m * Ncols + c0 + idx] = acc[r] + bv;
    }
  }
}

// ---------------- fold attn vectors into fc weights: w~[k][h] = sum_d W[k][h*64+d]*attn[h][d]
__global__ void fold_attn_kernel(const float* __restrict__ fcw,
                                 const float* __restrict__ fcew,
                                 const float* __restrict__ al,
                                 const float* __restrict__ ar,
                                 const float* __restrict__ ae,
                                 float* __restrict__ wl, float* __restrict__ wr,
                                 float* __restrict__ we) {
  int t = blockIdx.x * blockDim.x + threadIdx.x;
  if (t >= 3 * 64 * 4) return;
  int which = t >> 8;           // 0=l, 1=r, 2=e
  int k = (t & 255) >> 2, h = t & 3;
  const float* W  = (which == 2) ? fcew : fcw;
  const float* av = (which == 0) ? al : (which == 1) ? ar : ae;
  float s = 0.0f;
  for (int d = 0; d < 64; ++d) s += W[k * 256 + h * 64 + d] * av[h * 64 + d];
  float* out = (which == 0) ? wl : (which == 1) ? wr : we;
  out[k * 4 + h] = s;
}

// ---------------- el/er per node ----------------
__global__ void el_er_kernel(const float* __restrict__ nodes,
                             const float* __restrict__ wl,
                             const float* __restrict__ wr,
                             float* __restrict__ el, float* __restrict__ er, int N) {
  __shared__ float swl[256], swr[256];
  for (int i = threadIdx.x; i < 256; i += blockDim.x) { swl[i] = wl[i]; swr[i] = wr[i]; }
  __syncthreads();
  int n = blockIdx.x * blockDim.x + threadIdx.x;
  if (n >= N) return;
  const float* row = nodes + (size_t)n * 64;
  float l[4] = {0, 0, 0, 0}, r[4] = {0, 0, 0, 0};
  for (int k = 0; k < 64; ++k) {
    float v = row[k];
#pragma unroll
    for (int h = 0; h < 4; ++h) { l[h] += v * swl[k * 4 + h]; r[h] += v * swr[k * 4 + h]; }
  }
#pragma unroll
  for (int h = 0; h < 4; ++h) { el[n * 4 + h] = l[h]; er[n * 4 + h] = r[h]; }
}

// ---------------- edge logits: leaky_relu(el[src]+er[dst]+ee); atomic segment max
__global__ void edge_logit_kernel(const float* __restrict__ edgesP,
                                  const float* __restrict__ we,
                                  const float* __restrict__ el,
                                  const float* __restrict__ er,
                                  const int* __restrict__ src,
                                  const int* __restrict__ dst,
                                  float* __restrict__ elog,
                                  float* __restrict__ emax, int E) {
  __shared__ float swe[256];
  for (int i = threadIdx.x; i < 256; i += blockDim.x) swe[i] = we[i];
  __syncthreads();
  int e = blockIdx.x * blockDim.x + threadIdx.x;
  if (e >= E) return;
  const float* row = edgesP + (size_t)e * 64;
  float a[4] = {0, 0, 0, 0};
  for (int k = 0; k < 64; ++k) {
    float v = row[k];
#pragma unroll
    for (int h = 0; h < 4; ++h) a[h] += v * swe[k * 4 + h];
  }
  int s = src[e], d = dst[e];
#pragma unroll
  for (int h = 0; h < 4; ++h) {
    float lg = el[s * 4 + h] + er[d * 4 + h] + a[h];
    lg = lg > 0.0f ? lg : 0.2f * lg;
    elog[e * 4 + h] = lg;
    atomicMaxF(&emax[d * 4 + h], lg);
  }
}

// ---------------- ex = exp(e - emax[dst]); denom[dst] += ex ----------------
__global__ void exp_denom_kernel(float* __restrict__ elog,
                                 const float* __restrict__ emax,
                                 float* __restrict__ denom,
                                 const int* __restrict__ dst, int E) {
  int e = blockIdx.x * blockDim.x + threadIdx.x;
  if (e >= E) return;
  int d = dst[e];
#pragma unroll
  for (int h = 0; h < 4; ++h) {
    float x = expf(elog[e * 4 + h] - emax[d * 4 + h]);
    elog[e * 4 + h] = x;
    atomicAdd(&denom[d * 4 + h], x);
  }
}

// ---------------- fused: fe-tile via WMMA, softmax weight, gather ft[src], scatter to rst[dst]
__global__ void message_kernel(const float* __restrict__ edgesP,
                               const float* __restrict__ W,   // fc_edge_w[2] 64x256
                               const float* __restrict__ ft,
                               const float* __restrict__ ex,
                               const float* __restrict__ denom,
                               const int* __restrict__ src,
                               const int* __restrict__ dst,
                               float* __restrict__ rst, int E) {
  const int lane = threadIdx.x & 31;
  const int half = lane >> 4, idx = lane & 15;
  const int colTiles = 16;                       // 256/16
  const int totalTiles = (E >> 4) * colTiles;
  const int nWaves = (gridDim.x * blockDim.x) >> 5;
  for (int wave = (blockIdx.x * blockDim.x + threadIdx.x) >> 5; wave < totalTiles;
       wave += nWaves) {
    const int et = wave / colTiles, ct = wave % colTiles;
    const int row0 = et * 16, c0 = ct * 16;
    const int h = ct >> 2;                        // head index (16-col tile within one head)
    const float* arow = edgesP + (size_t)(row0 + idx) * 64;
    v8f acc = {};
#pragma unroll
    for (int k0 = 0; k0 < 64; k0 += 4) {
      const int ka = k0 + 2 * half;
      v2f a, b;
      a[0] = arow[ka];
      a[1] = arow[ka + 1];
      b[0] = W[(size_t)ka * 256 + c0 + idx];
      b[1] = W[(size_t)(ka + 1) * 256 + c0 + idx];
      acc = wmma4(a, b, acc);                     // acc += edges_tile @ W_tile  (fe tile)
    }
    const int col = c0 + idx;
#pragma unroll
    for (int r = 0; r < 8; ++r) {
      const int e = row0 + r + 8 * half;
      const int s = src[e], d = dst[e];
      const float ae = ex[e * 4 + h] / denom[d * 4 + h];
      const float val = ae * (ft[(size_t)s * 256 + col] + acc[r]);
      atomicAdd(&rst[(size_t)d * 256 + col], val);
    }
  }
}

// ---------------- SiLU(rst + bias) in-place + BN batch stats over (N,H) per channel D
__global__ void silu_bn_stats_kernel(float* __restrict__ rst,
                                     const float* __restrict__ cbias,
                                     float* __restrict__ bnsum, int total) {
  __shared__ float ssum[64], ssq[64];
  if (threadIdx.x < 64) { ssum[threadIdx.x] = 0.0f; ssq[threadIdx.x] = 0.0f; }
  __syncthreads();
  const int stride = gridDim.x * blockDim.x;      // blockDim==256 -> stride%256==0
  const int start = blockIdx.x * blockDim.x + threadIdx.x;
  const int c = start & 63;                       // channel fixed per thread
  const float bv = cbias[start & 255];
  float lsum = 0.0f, lsq = 0.0f;
  for (int i = start; i < total; i += stride) {
    float v = rst[i] + bv;
    float act = v / (1.0f + expf(-v));            // silu
    rst[i] = act;
    lsum += act; lsq += act * act;
  }
  atomicAdd(&ssum[c], lsum);
  atomicAdd(&ssq[c], lsq);
  __syncthreads();
  if (threadIdx.x < 64) {
    atomicAdd(&bnsum[threadIdx.x], ssum[threadIdx.x]);
    atomicAdd(&bnsum[64 + threadIdx.x], ssq[threadIdx.x]);
  }
}

__global__ void bn_finalize_kernel(const float* __restrict__ bnsum,
                                   const float* __restrict__ gamma,
                                   const float* __restrict__ beta,
                                   float* __restrict__ ss, float invcnt) {
  int c = blockIdx.x * blockDim.x + threadIdx.x;
  if (c >= 64) return;
  float mean = bnsum[c] * invcnt;
  float var = bnsum[64 + c] * invcnt - mean * mean;
  float sc = rsqrtf(var + 1e-5f) * gamma[c];
  ss[c] = sc;                                     // scale
  ss[64 + c] = beta[c] - mean * sc;               // shift
}

// ---------------- head: x = BN(act) @ head_w + head_b, affine fused into A loads
__global__ void head_kernel(const float* __restrict__ act,
                            const float* __restrict__ W,
                            const float* __restrict__ bias,
                            const float* __restrict__ ss,
                            float* __restrict__ x, int N) {
  __shared__ float sss[128];
  for (int i = threadIdx.x; i < 128; i += blockDim.x) sss[i] = ss[i];
  __syncthreads();
  const int lane = threadIdx.x & 31;
  const int half = lane >> 4, idx = lane & 15;
  const int totalTiles = (N >> 4) * 4;            // 4 col tiles (64 cols)
  const int nWaves = (gridDim.x * blockDim.x) >> 5;
  for (int wave = (blockIdx.x * blockDim.x + threadIdx.x) >> 5; wave < totalTiles;
       wave += nWaves) {
    const int rt = wave >> 2, ct = wave & 3;
    const int c0 = ct * 16;
    const float* arow = act + (size_t)(rt * 16 + idx) * 256;
    v8f acc = {};
    for (int k0 = 0; k0 < 256; k0 += 4) {
      const int ka = k0 + 2 * half;
      v2f a, b;
      a[0] = arow[ka]     * sss[ka & 63]       + sss[64 + (ka & 63)];
      a[1] = arow[ka + 1] * sss[(ka + 1) & 63] + sss[64 + ((ka + 1) & 63)];
      b[0] = W[(size_t)ka * 64 + c0 + idx];
      b[1] = W[(size_t)(ka + 1) * 64 + c0 + idx];
      acc = wmma4(a, b, acc);
    }
    const float bv = bias[c0 + idx];
#pragma unroll
    for (int r = 0; r < 8; ++r)
      x[(size_t)(rt * 16 + r + 8 * half) * 64 + c0 + idx] = acc[r] + bv;
  }
}

// ---------------- mean-pool accumulate ----------------
__global__ void pool_accum_kernel(const float* __restrict__ x,
                                  const int* __restrict__ gid,
                                  float* __restrict__ psum,
                                  float* __restrict__ pcnt, int N) {
  int t = blockIdx.x * blockDim.x + threadIdx.x;
  if (t >= N * 64) return;
  int n = t >> 6, c = t & 63;
  int g = gid[n];
  atomicAdd(&psum[g * 64 + c], x[t]);
  if (c == 0) atomicAdd(&pcnt[g], 1.0f);
}

// ---------------- write output: [pooled (G,64)] ++ [x[nodes_idx] (Q,64)] ----------------
__global__ void output_kernel(const float* __restrict__ psum,
                              const float* __restrict__ pcnt,
                              const float* __restrict__ x,
                              const int* __restrict__ nidx,
                              float* __restrict__ out, int G, int Q) {
  int t = blockIdx.x * blockDim.x + threadIdx.x;
  int total = (G + Q) * 64;
  if (t >= total) return;
  if (t < G * 64) {
    out[t] = psum[t] / fmaxf(pcnt[t >> 6], 1.0f);
  } else {
    int q = (t - G * 64) >> 6, c = t & 63;
    out[t] = x[(size_t)nidx[q] * 64 + c];
  }
}

// =======================================================================
extern "C" void kernel_launch(void* const* d_in, const int* in_sizes, int n_in,
                              void* d_out, int out_size, void* d_ws, size_t ws_size,
                              hipStream_t stream) {
  const float* node_feat = (const float*)d_in[0];
  const float* edge_feat = (const float*)d_in[1];
  const float* node_proj_w = (const float*)d_in[2];
  const float* node_proj_b = (const float*)d_in[3];
  const float* edge_proj_w = (const float*)d_in[4];
  const float* edge_proj_b = (const float*)d_in[5];
  const float* fc_w      = (const float*)d_in[6];
  const float* fc_edge_w = (const float*)d_in[7];
  const float* attn_l    = (const float*)d_in[8];
  const float* attn_r    = (const float*)d_in[9];
  const float* attn_e    = (const float*)d_in[10];
  const float* conv_bias = (const float*)d_in[11];
  const float* bn_gamma  = (const float*)d_in[12];
  const float* bn_beta   = (const float*)d_in[13];
  const float* head_w    = (const float*)d_in[14];
  const float* head_b    = (const float*)d_in[15];
  const int* src  = (const int*)d_in[16];
  const int* dst  = (const int*)d_in[17];
  const int* gid  = (const int*)d_in[18];
  const int* nidx = (const int*)d_in[19];

  const int N = in_sizes[0] / 64;     // 50000
  const int E = in_sizes[1] / 16;     // 800000
  const int Q = in_sizes[19];         // 4096
  const int G = out_size / 64 - Q;    // 32
  const int LAST = 2;                 // only the final layer affects the output

  // layer-2 parameter slices
  const float* fcW2  = fc_w      + (size_t)LAST * 64 * 256;
  const float* fceW2 = fc_edge_w + (size_t)LAST * 64 * 256;
  const float* al2 = attn_l + (size_t)LAST * 4 * 64;
  const float* ar2 = attn_r + (size_t)LAST * 4 * 64;
  const float* ae2 = attn_e + (size_t)LAST * 4 * 64;
  const float* cb2 = conv_bias + (size_t)LAST * 256;
  const float* bg2 = bn_gamma + (size_t)LAST * 64;
  const float* bb2 = bn_beta  + (size_t)LAST * 64;

  // ---- workspace carve-out ----
  char* w = (char*)d_ws;
  auto alloc = [&](size_t bytes) { void* p = (void*)w; w += (bytes + 255) & ~(size_t)255; return p; };
  float* nodes  = (float*)alloc((size_t)N * 64 * 4);
  float* edgesP = (float*)alloc((size_t)E * 64 * 4);
  float* ft     = (float*)alloc((size_t)N * 256 * 4);
  float* rst    = (float*)alloc((size_t)N * 256 * 4);
  float* xbuf   = (float*)alloc((size_t)N * 64 * 4);
  float* elog   = (float*)alloc((size_t)E * 4 * 4);
  float* emax   = (float*)alloc((size_t)N * 4 * 4);
  float* denom  = (float*)alloc((size_t)N * 4 * 4);
  float* el     = (float*)alloc((size_t)N * 4 * 4);
  float* er     = (float*)alloc((size_t)N * 4 * 4);
  float* wl     = (float*)alloc(64 * 4 * 4);
  float* wr     = (float*)alloc(64 * 4 * 4);
  float* we     = (float*)alloc(64 * 4 * 4);
  float* bnsum  = (float*)alloc(128 * 4);
  float* ssbuf  = (float*)alloc(128 * 4);
  float* psum   = (float*)alloc((size_t)G * 64 * 4);
  float* pcnt   = (float*)alloc((size_t)G * 4);

  // ---- init accumulators ----
  hipMemsetAsync(rst,   0, (size_t)N * 256 * 4, stream);
  hipMemsetAsync(denom, 0, (size_t)N * 4 * 4, stream);
  hipMemsetAsync(bnsum, 0, 128 * 4, stream);
  hipMemsetAsync(psum,  0, (size_t)G * 64 * 4, stream);
  hipMemsetAsync(pcnt,  0, (size_t)G * 4, stream);
  hipMemsetAsync(emax, 0xFF, (size_t)N * 4 * 4, stream);  // -NaN bits; replaced by first atomicMaxF

  const int B = 256;  // 8 wave32s per block

  // fold attention vectors into fc weights
  fold_attn_kernel<<<3, B, 0, stream>>>(fcW2, fceW2, al2, ar2, ae2, wl, wr, we);

  // projections + ft  (WMMA fp32 GEMMs)
  {
    int tiles = (N / 16) * (64 / 16);
    gemm_bias_kernel<<<(tiles + 7) / 8, B, 0, stream>>>(node_feat, node_proj_w,
                                                        node_proj_b, nodes, N, 64, 64);
  }
  {
    int tiles = (E / 16) * (64 / 16);
    gemm_bias_kernel<<<(tiles + 7) / 8, B, 0, stream>>>(edge_feat, edge_proj_w,
                                                        edge_proj_b, edgesP, E, 16, 64);
  }
  {
    int tiles = (N / 16) * (256 / 16);
    gemm_bias_kernel<<<(tiles + 7) / 8, B, 0, stream>>>(nodes, fcW2, nullptr, ft,
                                                        N, 64, 256);
  }

  // attention logits + edge softmax
  el_er_kernel<<<(N + B - 1) / B, B, 0, stream>>>(nodes, wl, wr, el, er, N);
  edge_logit_kernel<<<(E + B - 1) / B, B, 0, stream>>>(edgesP, we, el, er, src, dst,
                                                       elog, emax, E);
  exp_denom_kernel<<<(E + B - 1) / B, B, 0, stream>>>(elog, emax, denom, dst, E);

  // fused fe-GEMM + weighted message scatter
  {
    int tiles = (E / 16) * 16;
    message_kernel<<<(tiles + 7) / 8, B, 0, stream>>>(edgesP, fceW2, ft, elog, denom,
                                                      src, dst, rst, E);
  }

  // SiLU + BN
  silu_bn_stats_kernel<<<512, B, 0, stream>>>(rst, cb2, bnsum, N * 256);
  bn_finalize_kernel<<<1, 64, 0, stream>>>(bnsum, bg2, bb2, ssbuf,
                                           1.0f / (float)(N * 4));

  // head projection with fused BN affine
  {
    int tiles = (N / 16) * 4;
    head_kernel<<<(tiles + 7) / 8, B, 0, stream>>>(rst, head_w, head_b, ssbuf, xbuf, N);
  }

  // pooling + gather into d_out
  pool_accum_kernel<<<(N * 64 + B - 1) / B, B, 0, stream>>>(xbuf, gid, psum, pcnt, N);
  output_kernel<<<((G + Q) * 64 + B - 1) / B, B, 0, stream>>>(psum, pcnt, xbuf, nidx,
                                                              (float*)d_out, G, Q);
  (void)n_in; (void)ws_size;
}